// SinkhornDistance_58643483460103
// MI455X (gfx1250) — compile-verified
//
#include <hip/hip_runtime.h>

typedef __attribute__((ext_vector_type(2))) float v2f;
typedef __attribute__((ext_vector_type(8))) float v8f;

#define BATCH 16
#define L 1024
#define DDIM 256
#define EPS 0.1f
#define INV_EPS 10.0f
#define N_ITER 50
#define LOG_MU (-6.93147180559945f) /* -log(1024) */

__device__ __forceinline__ float wave_max(float v) {
#pragma unroll
  for (int off = 16; off > 0; off >>= 1) v = fmaxf(v, __shfl_xor(v, off, 32));
  return v;
}
__device__ __forceinline__ float wave_sum(float v) {
#pragma unroll
  for (int off = 16; off > 0; off >>= 1) v += __shfl_xor(v, off, 32);
  return v;
}

// ---------------- init: zero log_u, log_v, out --------------------------------
__global__ void k_init(float* __restrict__ logu, float* __restrict__ logv,
                       float* __restrict__ out) {
  int i = blockIdx.x * blockDim.x + threadIdx.x;
  if (i < BATCH * L) { logu[i] = 0.0f; logv[i] = 0.0f; }
  if (i == 0) out[0] = 0.0f;
}

// ---------------- squared norms of rows ---------------------------------------
__global__ void k_sqnorm(const float* __restrict__ x, const float* __restrict__ y,
                         float* __restrict__ x2, float* __restrict__ y2) {
  int wave = (blockIdx.x * blockDim.x + threadIdx.x) >> 5;
  int lane = threadIdx.x & 31;
  const float* src; float* dst; int row;
  if (wave < BATCH * L) { src = x; dst = x2; row = wave; }
  else                  { src = y; dst = y2; row = wave - BATCH * L; }
  const float4* p = (const float4*)(src + (size_t)row * DDIM);
  float s = 0.0f;
#pragma unroll
  for (int t = 0; t < 2; ++t) {
    float4 v = p[lane + 32 * t];
    s += v.x * v.x + v.y * v.y + v.z * v.z + v.w * v.w;
  }
  s = wave_sum(s);
  if (lane == 0) dst[row] = s;
}

// ---------------- cost + log_k via f32 WMMA -----------------------------------
// One wave computes a 16x16 tile of xy = x @ y^T, accumulating 64 x
// V_WMMA_F32_16X16X4_F32. A(16x4) layout: lane<16 -> M=lane, K={k,k+1};
// lane>=16 -> K={k+2,k+3}. B(4x16) symmetric with N=lane%16. C: lane%16 = N,
// vgpr r = M=r (lanes 0-15) / M=r+8 (lanes 16-31).
__global__ void k_cost(const float* __restrict__ x, const float* __restrict__ y,
                       const float* __restrict__ x2, const float* __restrict__ y2,
                       float* __restrict__ logk) {
  int wave = (blockIdx.x * blockDim.x + threadIdx.x) >> 5;
  int lane = threadIdx.x & 31;
  int hf  = lane >> 4;
  int idx = lane & 15;
  int b  = wave >> 12;          // / 4096 tiles per batch
  int t  = wave & 4095;
  int i0 = (t >> 6) << 4;
  int j0 = (t & 63) << 4;

  const float* xr = x + (size_t)(b * L + i0 + idx) * DDIM + 2 * hf;
  const float* yr = y + (size_t)(b * L + j0 + idx) * DDIM + 2 * hf;

  v8f c = {0.f, 0.f, 0.f, 0.f, 0.f, 0.f, 0.f, 0.f};
#pragma unroll 8
  for (int k = 0; k < DDIM; k += 4) {
    v2f a  = *(const v2f*)(xr + k);
    v2f bb = *(const v2f*)(yr + k);
    c = __builtin_amdgcn_wmma_f32_16x16x4_f32(false, a, false, bb,
                                              (short)0, c, false, false);
  }

  int j = j0 + idx;
  float y2v = y2[b * L + j];
#pragma unroll
  for (int r = 0; r < 8; ++r) {
    int i = i0 + r + 8 * hf;
    float d2 = x2[b * L + i] + y2v - 2.0f * c[r];
    float d  = sqrtf(fmaxf(d2, 0.0f));
    logk[((size_t)(b * L + i) << 10) + j] = -d * INV_EPS;
  }
}

// ---------------- row logsumexp: log_u update ---------------------------------
// One wave per (b,i) row; each lane holds 32 elements in registers,
// exact two-pass (max then exp-sum) with one exp per element.
__global__ void k_update_u(const float* __restrict__ logk,
                           const float* __restrict__ logv,
                           float* __restrict__ logu) {
  int wave = (blockIdx.x * blockDim.x + threadIdx.x) >> 5;  // = b*L + i
  int lane = threadIdx.x & 31;
  int b = wave >> 10;
  const float4* kp = (const float4*)(logk + ((size_t)wave << 10));
  const float4* vp = (const float4*)(logv + b * L);
  float vals[32];
#pragma unroll
  for (int t = 0; t < 8; ++t) {
    float4 kv = kp[t * 32 + lane];
    float4 vv = vp[t * 32 + lane];
    vals[4 * t + 0] = kv.x + vv.x;
    vals[4 * t + 1] = kv.y + vv.y;
    vals[4 * t + 2] = kv.z + vv.z;
    vals[4 * t + 3] = kv.w + vv.w;
  }
  float m = vals[0];
#pragma unroll
  for (int e = 1; e < 32; ++e) m = fmaxf(m, vals[e]);
  m = wave_max(m);
  float s = 0.0f;
#pragma unroll
  for (int e = 0; e < 32; ++e) s += __expf(vals[e] - m);
  s = wave_sum(s);
  if (lane == 0) logu[wave] = LOG_MU - (m + __logf(s));
}

// ---------------- column logsumexp: log_v update ------------------------------
// Block = 64 columns x 4 row-segments (256 rows each). Coalesced column walks,
// chunked (8-wide) online LSE (~1.125 exp/elem), LDS combine of partial (m,s).
__global__ void k_update_v(const float* __restrict__ logk,
                           const float* __restrict__ logu,
                           float* __restrict__ logv) {
  __shared__ float su[L];
  __shared__ float pm[256];
  __shared__ float ps[256];
  int b  = blockIdx.x >> 4;
  int j0 = (blockIdx.x & 15) << 6;
  int jj  = threadIdx.x & 63;
  int sub = threadIdx.x >> 6;     // 0..3 row segment
  int j = j0 + jj;
  for (int t = threadIdx.x; t < L; t += 256) su[t] = logu[b * L + t];
  __syncthreads();

  const float* kp = logk + ((size_t)b << 20) + j;
  float m = -3.0e38f, s = 0.0f;
  int ibase = sub << 8;
  for (int ii = 0; ii < 256; ii += 8) {
    int i = ibase + ii;
    float v[8];
#pragma unroll
    for (int e = 0; e < 8; ++e)
      v[e] = kp[(size_t)(i + e) << 10] + su[i + e];
    float cm = v[0];
#pragma unroll
    for (int e = 1; e < 8; ++e) cm = fmaxf(cm, v[e]);
    float nm = fmaxf(m, cm);
    s *= __expf(m - nm);
#pragma unroll
    for (int e = 0; e < 8; ++e) s += __expf(v[e] - nm);
    m = nm;
  }
  pm[threadIdx.x] = m;
  ps[threadIdx.x] = s;
  __syncthreads();
  if (threadIdx.x < 64) {
    float m0 = pm[threadIdx.x],       m1 = pm[threadIdx.x + 64];
    float m2 = pm[threadIdx.x + 128], m3 = pm[threadIdx.x + 192];
    float M = fmaxf(fmaxf(m0, m1), fmaxf(m2, m3));
    float S = ps[threadIdx.x]       * __expf(m0 - M)
            + ps[threadIdx.x + 64]  * __expf(m1 - M)
            + ps[threadIdx.x + 128] * __expf(m2 - M)
            + ps[threadIdx.x + 192] * __expf(m3 - M);
    logv[b * L + j0 + threadIdx.x] = LOG_MU - (M + __logf(S));
  }
}

// ---------------- final: sum(exp(lu + lk + lv) * cost), cost = -eps*lk --------
__global__ void k_final(const float* __restrict__ logk,
                        const float* __restrict__ logu,
                        const float* __restrict__ logv,
                        float* __restrict__ out) {
  __shared__ float wsum[8];
  int wave = (blockIdx.x * blockDim.x + threadIdx.x) >> 5;  // row id
  int lane = threadIdx.x & 31;
  int b = wave >> 10;
  float lu = logu[wave];
  const float4* kp = (const float4*)(logk + ((size_t)wave << 10));
  const float4* vp = (const float4*)(logv + b * L);
  float acc = 0.0f;
#pragma unroll
  for (int t = 0; t < 8; ++t) {
    float4 kv = kp[t * 32 + lane];
    float4 vv = vp[t * 32 + lane];
    acc += __expf(lu + kv.x + vv.x) * (-EPS * kv.x);
    acc += __expf(lu + kv.y + vv.y) * (-EPS * kv.y);
    acc += __expf(lu + kv.z + vv.z) * (-EPS * kv.z);
    acc += __expf(lu + kv.w + vv.w) * (-EPS * kv.w);
  }
  acc = wave_sum(acc);
  int w = threadIdx.x >> 5;
  if (lane == 0) wsum[w] = acc;
  __syncthreads();
  if (threadIdx.x == 0) {
    float s = 0.0f;
#pragma unroll
    for (int k = 0; k < 8; ++k) s += wsum[k];
    atomicAdd(out, s * (1.0f / BATCH));
  }
}

extern "C" void kernel_launch(void* const* d_in, const int* in_sizes, int n_in,
                              void* d_out, int out_size, void* d_ws, size_t ws_size,
                              hipStream_t stream) {
  const float* x = (const float*)d_in[0];
  const float* y = (const float*)d_in[1];
  float* out = (float*)d_out;

  float* ws   = (float*)d_ws;
  float* logk = ws;                                   // 16M floats = 64 MB
  float* x2   = ws + (size_t)BATCH * L * L;           // 16K
  float* y2   = x2 + BATCH * L;                       // 16K
  float* logu = y2 + BATCH * L;                       // 16K
  float* logv = logu + BATCH * L;                     // 16K

  k_init<<<64, 256, 0, stream>>>(logu, logv, out);
  k_sqnorm<<<4096, 256, 0, stream>>>(x, y, x2, y2);   // 2*16K rows, 8 waves/blk
  k_cost<<<8192, 256, 0, stream>>>(x, y, x2, y2, logk); // 65536 tiles, 8 waves/blk
  for (int it = 0; it < N_ITER; ++it) {
    k_update_u<<<2048, 256, 0, stream>>>(logk, logv, logu);
    k_update_v<<<256, 256, 0, stream>>>(logk, logu, logv);
  }
  k_final<<<2048, 256, 0, stream>>>(logk, logu, logv, out);
}